// GATModel_10960756540205
// MI455X (gfx1250) — compile-verified
//
#include <hip/hip_runtime.h>

#define NEG_SLOPE 0.2f

typedef __attribute__((ext_vector_type(2))) float v2f;
typedef __attribute__((ext_vector_type(8))) float v8f;

// ---------------- fill ----------------
__global__ void fill_f32(float* __restrict__ p, float v, long n) {
  long i = blockIdx.x * (long)blockDim.x + threadIdx.x;
  if (i < n) p[i] = v;
}

// ---------------- WMMA f32 GEMM: D[M,NOUT] = A[M,K] * B[K,NOUT] ----------------
// One wave per 16x16 output tile, V_WMMA_F32_16X16X4_F32 over K in steps of 4.
// A 16x4 layout: lanes 0-15 -> K=0,1 ; lanes 16-31 -> K=2,3 (row = lane&15).
// B 4x16 layout: symmetric (col = lane&15).
// C/D: VGPR r holds M=r (lanes 0-15) and M=r+8 (lanes 16-31), N = lane&15.
template <int K, int NOUT>
__global__ void wmma_gemm_f32(const float* __restrict__ A,
                              const float* __restrict__ B,
                              float* __restrict__ D, int M) {
  const int wave = threadIdx.x >> 5;   // n-tile within block
  const int lane = threadIdx.x & 31;
  const int m0 = blockIdx.x << 4;
  const int n0 = wave << 4;
  if (m0 >= M) return;                 // uniform across wave
  const int r15 = lane & 15;
  const int hi  = lane >> 4;           // 0 or 1
  const float* ap = A + (size_t)(m0 + r15) * K + 2 * hi;
  const float* bp = B + (size_t)(2 * hi) * NOUT + (n0 + r15);
  v8f acc = {};
  for (int kk = 0; kk < K; kk += 4) {
    v2f a, b;
    a.x = ap[kk];
    a.y = ap[kk + 1];
    b.x = bp[(size_t)kk * NOUT];
    b.y = bp[(size_t)(kk + 1) * NOUT];
    acc = __builtin_amdgcn_wmma_f32_16x16x4_f32(false, a, false, b,
                                                (short)0, acc, false, false);
  }
  float* dp = D + (size_t)(m0 + (hi << 3)) * NOUT + (n0 + r15);
#pragma unroll
  for (int r = 0; r < 8; ++r) dp[(size_t)r * NOUT] = acc[r];
}

// ---------------- per-(node,head) attention logits ----------------
// one wave32 per (node, head); ch-wide dot with a_src and a_dst simultaneously
__global__ void alpha_k(const float* __restrict__ h,
                        const float* __restrict__ a_src,
                        const float* __restrict__ a_dst,
                        float* __restrict__ as, float* __restrict__ ad,
                        int nodes, int heads, int ch) {
  int wave = (int)((blockIdx.x * (long)blockDim.x + threadIdx.x) >> 5);
  int lane = threadIdx.x & 31;
  int total = nodes * heads;
  if (wave >= total) return;
  int node = wave / heads;
  int hd = wave - node * heads;
  const float* hp  = h + ((size_t)node * heads + hd) * ch;
  const float* sp  = a_src + (size_t)hd * ch;
  const float* dp  = a_dst + (size_t)hd * ch;
  float s = 0.f, d = 0.f;
  for (int c = lane; c < ch; c += 32) {
    float v = hp[c];
    s = fmaf(v, sp[c], s);
    d = fmaf(v, dp[c], d);
  }
#pragma unroll
  for (int off = 16; off > 0; off >>= 1) {
    s += __shfl_down(s, off, 32);
    d += __shfl_down(d, off, 32);
  }
  if (lane == 0) { as[wave] = s; ad[wave] = d; }
}

// ---------------- edge pass 1: e = leaky(as[src]+ad[dst]); segment max ----------------
template <int HEADS>
__global__ void edge_max_k(const int* __restrict__ srcA, const int* __restrict__ dstA,
                           const float* __restrict__ as, const float* __restrict__ ad,
                           float* __restrict__ eo, float* __restrict__ mx,
                           int Ereal, long tot) {
  long gid = blockIdx.x * (long)blockDim.x + threadIdx.x;
  if (gid >= tot) return;
  int eid = (int)(gid / HEADS);
  int hd  = (int)(gid - (long)eid * HEADS);
  int s = (eid < Ereal) ? srcA[eid] : (eid - Ereal);   // self-loops appended
  int d = (eid < Ereal) ? dstA[eid] : (eid - Ereal);
  float e = as[(size_t)s * HEADS + hd] + ad[(size_t)d * HEADS + hd];
  e = (e > 0.f) ? e : NEG_SLOPE * e;
  eo[gid] = e;
  __hip_atomic_fetch_max(&mx[(size_t)d * HEADS + hd], e,
                         __ATOMIC_RELAXED, __HIP_MEMORY_SCOPE_AGENT);
}

// ---------------- edge pass 2: ex = exp(e - m[dst]); segment sum ----------------
template <int HEADS>
__global__ void edge_exp_k(const int* __restrict__ dstA,
                           float* __restrict__ eo,
                           const float* __restrict__ mx, float* __restrict__ dn,
                           int Ereal, long tot) {
  long gid = blockIdx.x * (long)blockDim.x + threadIdx.x;
  if (gid >= tot) return;
  int eid = (int)(gid / HEADS);
  int hd  = (int)(gid - (long)eid * HEADS);
  int d = (eid < Ereal) ? dstA[eid] : (eid - Ereal);
  float ex = __expf(eo[gid] - mx[(size_t)d * HEADS + hd]);
  eo[gid] = ex;
  atomicAdd(&dn[(size_t)d * HEADS + hd], ex);
}

// ---------------- tiny pass: inv_dn = 1/(dn + 1e-16) (per node, per head) --------
__global__ void inv_k(float* __restrict__ dn, long n) {
  long i = blockIdx.x * (long)blockDim.x + threadIdx.x;
  if (i < n) dn[i] = 1.f / (dn[i] + 1e-16f);
}

// ---------------- edge pass 3: out[dst] += h[src] * (ex * inv_dn[dst]) ----------
// thread = (edge, head, 4-channel chunk); 16 consecutive lanes share (edge,head)
template <int HEADS, int CH>
__global__ void edge_agg_k(const int* __restrict__ srcA, const int* __restrict__ dstA,
                           const float* __restrict__ ex, const float* __restrict__ inv_dn,
                           const float* __restrict__ h, float* __restrict__ out,
                           int Ereal, long tot) {
  constexpr int CHUNKS = CH / 4;
  long gid = blockIdx.x * (long)blockDim.x + threadIdx.x;
  if (gid >= tot) return;
  int ck  = (int)(gid % CHUNKS);
  long t  = gid / CHUNKS;
  int hd  = (int)(t % HEADS);
  int eid = (int)(t / HEADS);
  int s = (eid < Ereal) ? srcA[eid] : (eid - Ereal);
  int d = (eid < Ereal) ? dstA[eid] : (eid - Ereal);
  float alpha = ex[(size_t)eid * HEADS + hd] * inv_dn[(size_t)d * HEADS + hd];
  const float4 hv = *(const float4*)(h + ((size_t)s * HEADS + hd) * CH + ck * 4);
  float* op = out + ((size_t)d * HEADS + hd) * CH + ck * 4;
  atomicAdd(op + 0, hv.x * alpha);
  atomicAdd(op + 1, hv.y * alpha);
  atomicAdd(op + 2, hv.z * alpha);
  atomicAdd(op + 3, hv.w * alpha);
}

// ---------------- bias (+ optional relu), in place ----------------
__global__ void bias_act_k(float* __restrict__ x, const float* __restrict__ b,
                           long n, int fmask, int do_relu) {
  long i = blockIdx.x * (long)blockDim.x + threadIdx.x;
  if (i >= n) return;
  float v = x[i] + b[(int)i & fmask];
  if (do_relu) v = (v > 0.f) ? v : 0.f;
  x[i] = v;
}

static inline int cdiv_l(long a, long b) { return (int)((a + b - 1) / b); }

extern "C" void kernel_launch(void* const* d_in, const int* in_sizes, int n_in,
                              void* d_out, int out_size, void* d_ws, size_t ws_size,
                              hipStream_t stream) {
  const float* x      = (const float*)d_in[0];
  const int*   ei     = (const int*)d_in[1];
  const float* W1     = (const float*)d_in[2];
  const float* a_src1 = (const float*)d_in[3];
  const float* a_dst1 = (const float*)d_in[4];
  const float* b1     = (const float*)d_in[5];
  const float* W2     = (const float*)d_in[6];
  const float* a_src2 = (const float*)d_in[7];
  const float* a_dst2 = (const float*)d_in[8];
  const float* b2     = (const float*)d_in[9];
  float* out = (float*)d_out;

  const int Nn    = in_sizes[0] / 128;   // 50000
  const int Ereal = in_sizes[1] / 2;     // 1,600,000
  const int Etot  = Ereal + Nn;
  const int* srcA = ei;
  const int* dstA = ei + Ereal;

  // workspace carve (256B aligned slices)
  char* wsb = (char*)d_ws;
  auto alloc = [&](size_t nf) {
    float* p = (float*)wsb;
    wsb += ((nf * sizeof(float) + 255) & ~(size_t)255);
    return p;
  };
  float* h1   = alloc((size_t)Nn * 256);
  float* agg1 = alloc((size_t)Nn * 256);
  float* h2   = alloc((size_t)Nn * 64);
  float* as1  = alloc((size_t)Nn * 4);
  float* ad1  = alloc((size_t)Nn * 4);
  float* m1   = alloc((size_t)Nn * 4);
  float* dn1  = alloc((size_t)Nn * 4);
  float* as2  = alloc((size_t)Nn);
  float* ad2  = alloc((size_t)Nn);
  float* m2   = alloc((size_t)Nn);
  float* dn2  = alloc((size_t)Nn);
  float* e1   = alloc((size_t)Etot * 4);
  float* e2   = alloc((size_t)Etot);
  (void)ws_size; (void)n_in; (void)out_size;

  const float NEG_INF = -__builtin_inff();

  // ---------------- Layer 1 (H=4, C=64) ----------------
  wmma_gemm_f32<128, 256><<<cdiv_l(Nn, 16), 512, 0, stream>>>(x, W1, h1, Nn);

  alpha_k<<<cdiv_l((long)Nn * 4 * 32, 256), 256, 0, stream>>>(
      h1, a_src1, a_dst1, as1, ad1, Nn, 4, 64);

  fill_f32<<<cdiv_l((long)Nn * 4, 256), 256, 0, stream>>>(m1, NEG_INF, (long)Nn * 4);
  fill_f32<<<cdiv_l((long)Nn * 4, 256), 256, 0, stream>>>(dn1, 0.f, (long)Nn * 4);
  fill_f32<<<cdiv_l((long)Nn * 256, 256), 256, 0, stream>>>(agg1, 0.f, (long)Nn * 256);

  const long tot1 = (long)Etot * 4;
  edge_max_k<4><<<cdiv_l(tot1, 256), 256, 0, stream>>>(srcA, dstA, as1, ad1, e1, m1, Ereal, tot1);
  edge_exp_k<4><<<cdiv_l(tot1, 256), 256, 0, stream>>>(dstA, e1, m1, dn1, Ereal, tot1);
  inv_k<<<cdiv_l((long)Nn * 4, 256), 256, 0, stream>>>(dn1, (long)Nn * 4);
  const long tota1 = tot1 * 16;  // 4-channel chunks of 64
  edge_agg_k<4, 64><<<cdiv_l(tota1, 256), 256, 0, stream>>>(
      srcA, dstA, e1, dn1, h1, agg1, Ereal, tota1);

  bias_act_k<<<cdiv_l((long)Nn * 256, 256), 256, 0, stream>>>(
      agg1, b1, (long)Nn * 256, 255, 1);

  // ---------------- Layer 2 (H=1, C=64) ----------------
  wmma_gemm_f32<256, 64><<<cdiv_l(Nn, 16), 128, 0, stream>>>(agg1, W2, h2, Nn);

  alpha_k<<<cdiv_l((long)Nn * 32, 256), 256, 0, stream>>>(
      h2, a_src2, a_dst2, as2, ad2, Nn, 1, 64);

  fill_f32<<<cdiv_l((long)Nn, 256), 256, 0, stream>>>(m2, NEG_INF, (long)Nn);
  fill_f32<<<cdiv_l((long)Nn, 256), 256, 0, stream>>>(dn2, 0.f, (long)Nn);
  fill_f32<<<cdiv_l((long)Nn * 64, 256), 256, 0, stream>>>(out, 0.f, (long)Nn * 64);

  const long tot2 = (long)Etot;
  edge_max_k<1><<<cdiv_l(tot2, 256), 256, 0, stream>>>(srcA, dstA, as2, ad2, e2, m2, Ereal, tot2);
  edge_exp_k<1><<<cdiv_l(tot2, 256), 256, 0, stream>>>(dstA, e2, m2, dn2, Ereal, tot2);
  inv_k<<<cdiv_l((long)Nn, 256), 256, 0, stream>>>(dn2, (long)Nn);
  const long tota2 = tot2 * 16;
  edge_agg_k<1, 64><<<cdiv_l(tota2, 256), 256, 0, stream>>>(
      srcA, dstA, e2, dn2, h2, out, Ereal, tota2);

  bias_act_k<<<cdiv_l((long)Nn * 64, 256), 256, 0, stream>>>(
      out, b2, (long)Nn * 64, 63, 0);
}